// MotionEstimation_55559696941391
// MI455X (gfx1250) — compile-verified
//
#include <hip/hip_runtime.h>

// MLS warp, restructured as a per-batch moment GEMM:
//   D(1024x16) = W(1024x68) @ F(68x16)   via V_WMMA_F32_16X16X4_F32
// followed by a tiny per-pixel 2x2 solve.
//
// Round-2 change: B-matrix columns and control-point coords are hoisted out of
// the pixel-tile loop into registers (they are tile-invariant), so the hot loop
// is pure  v_rcp / v_pk_fma  ->  v_wmma  with no repeated LDS reads.

#define GRID 32
#define NCTRL 68
#define NMOM 16      // 12 moments used, padded to 16
#define KSTEPS 17    // 68 / 4

typedef __attribute__((ext_vector_type(2))) float v2f;
typedef __attribute__((ext_vector_type(8))) float v8f;

__global__ __launch_bounds__(256)
void mls_moment_wmma_kernel(const float* __restrict__ sp,
                            const float* __restrict__ dp,
                            float* __restrict__ out)
{
    __shared__ float pfx[NCTRL];
    __shared__ float pfy[NCTRL];
    __shared__ float Fm[NCTRL][NMOM];   // B-matrix: row k (control), col n (moment)
    __shared__ float Dt[8][16][17];     // per-wave D tile, padded to dodge bank conflicts

    const int b    = blockIdx.x;
    const int tid  = threadIdx.x;
    const int lane = tid & 31;
    const int wave = tid >> 5;

    // ---- setup: control-point table (coord-swapped, int16-truncated as in ref) ----
    if (tid < NCTRL) {
        const int base = (b * NCTRL + tid) * 2;
        const float px = (float)(short)dp[base + 1];
        const float py = (float)(short)dp[base + 0];
        const float qx = (float)(short)sp[base + 1];
        const float qy = (float)(short)sp[base + 0];
        pfx[tid] = px; pfy[tid] = py;
        Fm[tid][0]  = 1.0f;
        Fm[tid][1]  = px;
        Fm[tid][2]  = py;
        Fm[tid][3]  = px * px;
        Fm[tid][4]  = px * py;
        Fm[tid][5]  = py * py;
        Fm[tid][6]  = qx;
        Fm[tid][7]  = qy;
        Fm[tid][8]  = px * qx;
        Fm[tid][9]  = px * qy;
        Fm[tid][10] = py * qx;
        Fm[tid][11] = py * qy;
        Fm[tid][12] = 0.0f; Fm[tid][13] = 0.0f; Fm[tid][14] = 0.0f; Fm[tid][15] = 0.0f;
    }
    __syncthreads();

    const int half = lane >> 4;   // K half-split of the wave (ISA 16x4 f32 A/B layout)
    const int ln   = lane & 15;

    // ---- hoist tile-invariant operands into registers (read LDS once) ----
    v2f Breg[KSTEPS];   // B operand: F rows (kb, kb+1), column ln
    v2f Pxr[KSTEPS];    // pfx[kb], pfx[kb+1]  for this lane's K slots
    v2f Pyr[KSTEPS];    // pfy[kb], pfy[kb+1]
    #pragma unroll
    for (int ks = 0; ks < KSTEPS; ++ks) {
        const int kb = ks * 4 + half * 2;
        Breg[ks].x = Fm[kb][ln];
        Breg[ks].y = Fm[kb + 1][ln];
        Pxr[ks].x  = pfx[kb];
        Pxr[ks].y  = pfx[kb + 1];
        Pyr[ks].x  = pfy[kb];
        Pyr[ks].y  = pfy[kb + 1];
    }

    // ---- 8 tiles of 16 pixels per wave: 64 tiles cover all 1024 pixels ----
    for (int t = 0; t < 8; ++t) {
        const int tile = wave * 8 + t;
        const int pix0 = tile * 16;

        // This lane's A-matrix row pixel (M = lane&15)
        const int   apix = pix0 + ln;
        const float avx  = (float)(apix & 31);
        const float avy  = (float)(apix >> 5);

        v8f acc = {};
        #pragma unroll
        for (int ks = 0; ks < KSTEPS; ++ks) {
            // A elements: unnormalized MLS weights, synthesized in registers
            const float dx0 = Pxr[ks].x - avx, dy0 = Pyr[ks].x - avy;
            const float dx1 = Pxr[ks].y - avx, dy1 = Pyr[ks].y - avy;
            v2f a;
            a.x = __builtin_amdgcn_rcpf(dx0 * dx0 + dy0 * dy0 + 1e-8f);
            a.y = __builtin_amdgcn_rcpf(dx1 * dx1 + dy1 * dy1 + 1e-8f);

            // D = A x B + C  (16x16x4 f32)
            acc = __builtin_amdgcn_wmma_f32_16x16x4_f32(
                /*neg_a=*/false, a, /*neg_b=*/false, Breg[ks],
                /*c_mod=*/(short)0, acc, /*reuse_a=*/false, /*reuse_b=*/false);
        }

        // Spill D tile: lane holds rows M = v + 8*half, col N = ln
        #pragma unroll
        for (int v = 0; v < 8; ++v)
            Dt[wave][v + 8 * half][ln] = acc[v];
        __syncthreads();

        // ---- per-pixel finish: lanes 0..15 each own one pixel of the tile ----
        if (lane < 16) {
            const int   p   = pix0 + lane;
            const float pvx = (float)(p & 31);
            const float pvy = (float)(p >> 5);
            const float* d  = &Dt[wave][lane][0];

            const float inv0 = __builtin_amdgcn_rcpf(d[0]);   // 1 / m0
            const float psx  = d[1] * inv0, psy = d[2] * inv0;
            const float Pxx  = d[3] * inv0 - psx * psx;       // pTwp = E[p p^T] - pstar pstar^T
            const float Pxy  = d[4] * inv0 - psx * psy;
            const float Pyy  = d[5] * inv0 - psy * psy;
            const float qsx  = d[6] * inv0, qsy = d[7] * inv0;
            const float Bxx  = d[8]  * inv0 - psx * qsx;      // Sum w phat q^T
            const float Bxy  = d[9]  * inv0 - psx * qsy;
            const float Byx  = d[10] * inv0 - psy * qsx;
            const float Byy  = d[11] * inv0 - psy * qsy;

            const float det  = Pxx * Pyy - Pxy * Pxy;
            const float idet = __builtin_amdgcn_rcpf(det);
            const float ux   = pvx - psx, uy = pvy - psy;
            const float rx   = (ux * Pyy - uy * Pxy) * idet;  // u^T inv(pTwp)
            const float ry   = (uy * Pxx - ux * Pxy) * idet;

            float tx = rx * Bxx + ry * Byx + qsx;
            float ty = rx * Bxy + ry * Byy + qsy;

            // out-of-range pixels zeroed (not clipped), matching the reference
            tx = (tx < 0.0f || tx > (float)(GRID - 1)) ? 0.0f : tx;
            ty = (ty < 0.0f || ty > (float)(GRID - 1)) ? 0.0f : ty;

            out[(b * 2 + 0) * 1024 + p] = tx;
            out[(b * 2 + 1) * 1024 + p] = ty;
        }
        __syncthreads();
    }
}

extern "C" void kernel_launch(void* const* d_in, const int* in_sizes, int n_in,
                              void* d_out, int out_size, void* d_ws, size_t ws_size,
                              hipStream_t stream) {
    const float* sp = (const float*)d_in[0];
    const float* dp = (const float*)d_in[1];
    float* out = (float*)d_out;
    const int nB = in_sizes[0] / (NCTRL * 2);   // 512
    mls_moment_wmma_kernel<<<nB, 256, 0, stream>>>(sp, dp, out);
}